// Distance_52063593562436
// MI455X (gfx1250) — compile-verified
//
#include <hip/hip_runtime.h>
#include <math.h>

typedef __attribute__((ext_vector_type(2))) float v2f;
typedef __attribute__((ext_vector_type(8))) float v8f;

#define ATOMS 64
#define KNB   32
#define CUT2  25.0f

// bits(25.0f) = 0x41C80000 ; valid (d2 <= 25) <=> key < ((bits25+1) << 6)
#define KEY_LIMIT (((unsigned long long)0x41C80001ull) << 6)

// One workgroup (128 threads = 4 wave32) per molecule.
// Phase 1: 64x64 d2 matrix via V_WMMA_F32_16X16X4_F32; epilogue packs each
//          element into a sortable 64-bit key ((bits(d2)<<6)|j) in LDS.
// Phase 2: per-atom stable top-32 selection on keys + edge emission.
__launch_bounds__(128)
__global__ void radius_graph_wmma_kernel(const float* __restrict__ pos,
                                         float* __restrict__ out,
                                         int B) {
    __shared__ float spos[ATOMS][4];                    // x,y,z,|p|^2 (AoS)
    __shared__ unsigned long long keys[ATOMS * ATOMS];  // 32 KB

    const int mol  = blockIdx.x;
    const int tid  = threadIdx.x;          // 0..127
    const int wave = tid >> 5;             // 0..3 -> 16-row band
    const int lane = tid & 31;
    const int half = lane >> 4;            // 0: lanes 0-15, 1: lanes 16-31
    const int l16  = lane & 15;

    // ---- load this molecule's positions into LDS (AoS, 16B per atom) ----
    if (tid < ATOMS) {
        const float* p = pos + ((long long)mol * ATOMS + tid) * 3;
        float x = p[0], y = p[1], z = p[2];
        spos[tid][0] = x;
        spos[tid][1] = y;
        spos[tid][2] = z;
        spos[tid][3] = x * x + y * y + z * z;
        // warm L2/WGP$ for the next molecule's positions (global_prefetch_b8)
        __builtin_prefetch(p + 3LL * ATOMS, 0, 1);
    }
    __syncthreads();

    // ---- Phase 1: d2 = A(16x4) x B(4x16) + C via WMMA ----
    // A row i = (x, y, z, |p|^2); B col j = (-2x, -2y, -2z, 1); C = |p_j|^2.
    // A layout: lanes 0-15 hold (K0,K1), lanes 16-31 hold (K2,K3), M = lane&15.
    const int mrow = wave * 16 + l16;
    const v2f a = *(const v2f*)&spos[mrow][2 * half];   // (x,y) or (z,sq)

    #pragma unroll
    for (int tn = 0; tn < 4; ++tn) {
        const int ncol = tn * 16 + l16;
        const v2f pb = *(const v2f*)&spos[ncol][2 * half];
        v2f b;
        b.x = -2.0f * pb.x;                       // -2x  or -2z
        b.y = half ? 1.0f : (-2.0f * pb.y);       //  1   or -2y   (cndmask)

        const float sqc = spos[ncol][3];          // depends only on N
        v8f c;
        #pragma unroll
        for (int r = 0; r < 8; ++r) c[r] = sqc;

        v8f d = __builtin_amdgcn_wmma_f32_16x16x4_f32(
            /*neg_a=*/false, a, /*neg_b=*/false, b,
            /*c_mod=*/(short)0, c, /*reuse_a=*/false, /*reuse_b=*/false);

        // C/D layout: VGPR r -> M = r (lanes 0-15) or 8+r (lanes 16-31)
        // Pack sortable key: (bits(max(d2,0)) << 6) | j  (min key = nearest,
        // ties toward smaller j -> matches top_k stability).
        #pragma unroll
        for (int r = 0; r < 8; ++r) {
            const int row = wave * 16 + half * 8 + r;
            unsigned ub = __float_as_uint(d[r]);
            ub = (ub & 0x80000000u) ? 0u : ub;    // clamp -eps/-0 to +0
            keys[row * ATOMS + ncol] =
                ((unsigned long long)ub << 6) | (unsigned)ncol;
        }
    }
    __syncthreads();

    // ---- Phase 2: per-atom stable top-32 within radius; emit edges ----
    if (tid < ATOMS) {
        const int i = tid;
        const float xi = spos[i][0], yi = spos[i][1], zi = spos[i][2];
        const ulonglong2* row = (const ulonglong2*)&keys[i * ATOMS];
        unsigned long long used = 0ULL;

        const long long E  = (long long)B * ATOMS * KNB;
        float* srcf = out;                 // edge_index[0] as float
        float* dstf = out + E;             // edge_index[1] as float
        float* wgtf = out + 2 * E;         // edge_weight
        float* vecf = out + 3 * E;         // edge_vec [E,3]

        const long long ebase = ((long long)mol * ATOMS + i) * KNB;
        const int dst = mol * ATOMS + i;

        for (int k = 0; k < KNB; ++k) {
            unsigned long long best = ~0ULL;
            for (int jc = 0; jc < ATOMS / 2; ++jc) {
                const ulonglong2 kk = row[jc];           // ds_load_b128
                const int j0 = 2 * jc, j1 = 2 * jc + 1;
                if ((kk.x < KEY_LIMIT) && !((used >> j0) & 1ULL) && kk.x < best)
                    best = kk.x;
                if ((kk.y < KEY_LIMIT) && !((used >> j1) & 1ULL) && kk.y < best)
                    best = kk.y;
            }

            int nbr;
            if (best == ~0ULL) {
                nbr = i;                          // pad with self -> zero edge
            } else {
                nbr = (int)(best & 63ULL);
                used |= 1ULL << nbr;
            }

            const int src = mol * ATOMS + nbr;
            const float vx = spos[nbr][0] - xi;
            const float vy = spos[nbr][1] - yi;
            const float vz = spos[nbr][2] - zi;
            const float sqn = vx * vx + vy * vy + vz * vz;
            const float w = (src != dst) ? sqrtf(sqn) : 0.0f;

            const long long e = ebase + k;
            srcf[e] = (float)src;
            dstf[e] = (float)dst;
            wgtf[e] = w;
            vecf[3 * e + 0] = vx;
            vecf[3 * e + 1] = vy;
            vecf[3 * e + 2] = vz;
        }
    }
}

extern "C" void kernel_launch(void* const* d_in, const int* in_sizes, int n_in,
                              void* d_out, int out_size, void* d_ws, size_t ws_size,
                              hipStream_t stream) {
    const float* pos = (const float*)d_in[0];      // [N,3] float32
    // inputs: pos (N*3), batch (N), num_graphs (1), atoms_per_graph (1)
    const int N = in_sizes[1];                     // total atoms
    const int B = N / ATOMS;                       // molecules
    float* out = (float*)d_out;

    radius_graph_wmma_kernel<<<dim3(B), dim3(128), 0, stream>>>(pos, out, B);
}